// SPUTransformer_44358422233750
// MI455X (gfx1250) — compile-verified
//
#include <hip/hip_runtime.h>
#include <math.h>

// SPU transformer bounds kernel for MI455X (gfx1250).
// Streaming elementwise workload: 32 B HBM traffic/row -> ~11 us floor at
// 23.3 TB/s, with VALU cost of the same order. Strategy: fully branchless
// per-lane math (v_cndmask, no exec-mask churn), one b128 NT load + three
// b128 NT stores per thread (2 rows), global_prefetch_b8 read-ahead.
// No matrix structure -> WMMA inapplicable.

typedef float v4f __attribute__((ext_vector_type(4)));

#define SPU_EPS 1e-5f

// spu(x) and der_spu(x), branchless.
// spu(x)     = x>=0 ? x*x - 0.5 : sigmoid(-x) - 1
// der_spu(x) = x>=0 ? 2x        : -s*(1-s),  s = sigmoid(-x)
__device__ __forceinline__ void spu_der(float x, float& v, float& d)
{
    const float s  = 1.0f / (1.0f + expf(x));   // sigmoid(-x); overflow -> 0
    const bool  ge = (x >= 0.0f);
    v = ge ? fmaf(x, x, -0.5f) : (s - 1.0f);
    d = ge ? (x + x)           : (-s * (1.0f - s));
}

__device__ __forceinline__ void spu_process(float l, float u, float r[6])
{
    float vs_l, tan_l, vs_u, tan_u;
    spu_der(l, vs_l, tan_l);
    spu_der(u, vs_u, tan_u);

    const bool neg   = (u < 0.0f);          // from ORIGINAL u
    const bool pos   = (l >= 0.0f);
    const bool cross = !(neg || pos);

    const float all_slopes = (vs_u - vs_l) / (u - l);
    const bool cross_neg = (all_slopes < 0.0f) && cross;
    const bool cin_cross = (all_slopes < tan_l) && cross_neg;

    const float slope_u = cin_cross ? tan_l : all_slopes;
    const float shift_u = vs_l - slope_u * l;

    // Rare path: clip u to the tangent/parabola intersection. Keep as the
    // single divergent branch so waves with no cin_cross lane skip the
    // extra expf entirely.
    if (__builtin_expect(__any(cin_cross), 0)) {
        if (cin_cross) {
            const float arg = fmaf(slope_u, slope_u, 4.0f * (shift_u + 0.5f));
            u = 0.5f * (slope_u + sqrtf(arg));
            spu_der(u, vs_u, tan_u);        // refresh vs_u / tan_u
        }
    }

    const float safe_den = cross ? -l : 1.0f;
    const float pls_l  = cross ? ((-0.5f - vs_l) / safe_den) : tan_l;
    const float pls_u  = tan_u;
    const float spls_l = vs_l - pls_l * l;
    const float spls_u = vs_u - pls_u * u;
    const float cls    = cross ? -0.5f : vs_l;
    const float y_u    = pls_l * u + spls_l;
    const float y_l    = pls_u * l + spls_u;
    const float x      = (cls - shift_u) / slope_u;

    const bool m_hi = (x >= u - SPU_EPS);
    const bool m_lo = (x <= l + SPU_EPS);

    const float ul = u - l;
    const float a0 = 0.5f * fabsf(y_u - vs_u) * ul;
    const float a1 = 0.5f * fabsf(y_l - vs_l) * ul;
    float a2 = m_lo ? (0.5f * fabsf(vs_u - cls) * (u - x))
             : (m_hi ? (0.5f * fabsf(vs_l - cls) * (x - l)) : 0.0f);
    const float cutoff = m_lo ? (0.5f * fabsf(vs_l - cls) * (l - x))
                       : (m_hi ? (0.5f * fabsf(vs_u - cls) * (x - u)) : 0.0f);
    a2 = m_hi ? (a2 - cutoff) : a2;         // sequential, matches the wheres
    a2 = m_lo ? (a2 - cutoff) : a2;

    const bool switched = (all_slopes < 0.0f);   // ORIGINAL all_slopes
    float out_l = switched ? vs_u : vs_l;
    const float out_u = switched ? vs_l : vs_u;
    out_l = cross ? -0.5f : out_l;

    // argmin over {a0,a1,a2}, first occurrence wins ties (branchless)
    const bool pick1 = (a1 < a0);
    const float best01 = pick1 ? a1 : a0;
    const bool pick2 = (a2 < best01);
    const float slope_l = pick2 ? 0.0f : (pick1 ? pls_u  : pls_l);
    const float shift_l = pick2 ? cls  : (pick1 ? spls_u : spls_l);

    r[0] = out_l;
    r[1] = out_u;
    r[2] = neg ? slope_u : slope_l;
    r[3] = neg ? slope_l : slope_u;
    r[4] = neg ? shift_u : shift_l;
    r[5] = neg ? shift_l : shift_u;
}

// Hot path: exactly 2 rows per thread, straight-line code.
// One 16B NT load, three 16B NT stores, all 16B-aligned & coalesced.
__global__ __launch_bounds__(256) void spu_pair_kernel(
    const float* __restrict__ bounds,   // [n, 2] row-major
    float* __restrict__ out,            // [n, 6] row-major
    int npairs)
{
    const int p = blockIdx.x * blockDim.x + threadIdx.x;
    if (p >= npairs) return;

    // Run the read stream ahead of demand -> global_prefetch_b8
    // (speculative prefetch past the mapping is silently dropped).
    __builtin_prefetch(bounds + 4 * (size_t)p + 4096);

    // {l0, u0, l1, u1}
    const v4f b = __builtin_nontemporal_load((const v4f*)bounds + p);

    float r0[6], r1[6];
    spu_process(b.x, b.y, r0);
    spu_process(b.z, b.w, r1);

    v4f o0 = { r0[0], r0[1], r0[2], r0[3] };
    v4f o1 = { r0[4], r0[5], r1[0], r1[1] };
    v4f o2 = { r1[2], r1[3], r1[4], r1[5] };

    v4f* op = (v4f*)out + 3 * (size_t)p;    // 48-byte window per thread
    __builtin_nontemporal_store(o0, op + 0);
    __builtin_nontemporal_store(o1, op + 1);
    __builtin_nontemporal_store(o2, op + 2);
}

// Cold path: last row when n is odd (never launched for N = 8388608).
__global__ void spu_tail_kernel(const float* __restrict__ bounds,
                                float* __restrict__ out, int e)
{
    if (blockIdx.x == 0 && threadIdx.x == 0) {
        float r[6];
        spu_process(bounds[2 * (size_t)e], bounds[2 * (size_t)e + 1], r);
        #pragma unroll
        for (int k = 0; k < 6; ++k) out[6 * (size_t)e + k] = r[k];
    }
}

extern "C" void kernel_launch(void* const* d_in, const int* in_sizes, int n_in,
                              void* d_out, int out_size, void* d_ws, size_t ws_size,
                              hipStream_t stream)
{
    (void)n_in; (void)out_size; (void)d_ws; (void)ws_size;

    const float* bounds = (const float*)d_in[0];
    float* out = (float*)d_out;

    const int n      = in_sizes[0] / 2;     // number of (l,u) rows
    const int npairs = n / 2;               // 2 rows per thread

    if (npairs > 0) {
        const int threads = 256;            // 8 wave32 per block
        const int blocks  = (npairs + threads - 1) / threads;
        spu_pair_kernel<<<blocks, threads, 0, stream>>>(bounds, out, npairs);
    }
    if (n & 1) {
        spu_tail_kernel<<<1, 64, 0, stream>>>(bounds, out, n - 1);
    }
}